// DNDLSTM_47218870453063
// MI455X (gfx1250) — compile-verified
//
#include <hip/hip_runtime.h>
#include <math.h>

#define Hn   256
#define Dn   256
#define NGn  4
#define ROWS ((NGn + 1) * Hn)   // 1280
#define DICT 100000
#define OUTS 8
#define EPSF 1e-8f

typedef __attribute__((ext_vector_type(2))) float v2f;
typedef __attribute__((ext_vector_type(4))) float v4f;
typedef __attribute__((ext_vector_type(8))) float v8f;

__device__ __forceinline__ float waveReduceAdd(float v) {
#pragma unroll
  for (int m = 1; m < 32; m <<= 1) v += __shfl_xor(v, m, 32);
  return v;
}

__device__ __forceinline__ float sigmoidf_(float x) { return 1.0f / (1.0f + expf(-x)); }

// ---------------------------------------------------------------------------
// Zero the cross-block accumulators (must run every call: atomics accumulate).
// ---------------------------------------------------------------------------
__global__ void dnd_init(float* __restrict__ Vacc, float* __restrict__ Sacc) {
  int t = blockIdx.x * blockDim.x + threadIdx.x;
  if (t < Hn) Vacc[t] = 0.0f;
  if (t == 0) *Sacc = 0.0f;
}

// ---------------------------------------------------------------------------
// Gate pre-activations via V_WMMA_F32_16X16X4_F32 (full f32 precision).
// One wave per 16-row output tile. A = W tile (16x4), B = input vector
// replicated across all 16 N-columns, C chained (2 chains to hide latency).
//   preact[j] = dot(Wi[j],x) + dot(Wh[j],h) + bi[j] + bh[j]
// A layout (ISA 7.12.2, 32-bit 16x4): lanes 0-15: M=lane, K=(0,1) in v0,v1;
// lanes 16-31: M=lane-16, K=(2,3). B follows the same K pairing.
// D layout: lane 0 -> N=0, M=r (VGPR r); lane 16 -> N=0, M=r+8.
// ---------------------------------------------------------------------------
__global__ __launch_bounds__(32)
void dnd_gates_wmma(const float* __restrict__ x,  const float* __restrict__ h,
                    const float* __restrict__ Wi, const float* __restrict__ bi,
                    const float* __restrict__ Wh, const float* __restrict__ bh,
                    float* __restrict__ preact) {
  const int lane = threadIdx.x;
  const int row0 = blockIdx.x * 16;
  const int m    = lane & 15;
  const int ko   = (lane >> 4) << 1;  // 0 or 2

  v8f acc0 = {0.f, 0.f, 0.f, 0.f, 0.f, 0.f, 0.f, 0.f};
  v8f acc1 = {0.f, 0.f, 0.f, 0.f, 0.f, 0.f, 0.f, 0.f};

  const float* wrow = Wi + (size_t)(row0 + m) * Dn;
  for (int kk = 0; kk < Dn; kk += 8) {
    v2f a, b;
    a.x = wrow[kk + ko];     a.y = wrow[kk + ko + 1];
    b.x = x[kk + ko];        b.y = x[kk + ko + 1];
    acc0 = __builtin_amdgcn_wmma_f32_16x16x4_f32(false, a, false, b, (short)0, acc0, false, false);
    a.x = wrow[kk + 4 + ko]; a.y = wrow[kk + 4 + ko + 1];
    b.x = x[kk + 4 + ko];    b.y = x[kk + 4 + ko + 1];
    acc1 = __builtin_amdgcn_wmma_f32_16x16x4_f32(false, a, false, b, (short)0, acc1, false, false);
  }
  wrow = Wh + (size_t)(row0 + m) * Hn;
  for (int kk = 0; kk < Hn; kk += 8) {
    v2f a, b;
    a.x = wrow[kk + ko];     a.y = wrow[kk + ko + 1];
    b.x = h[kk + ko];        b.y = h[kk + ko + 1];
    acc0 = __builtin_amdgcn_wmma_f32_16x16x4_f32(false, a, false, b, (short)0, acc0, false, false);
    a.x = wrow[kk + 4 + ko]; a.y = wrow[kk + 4 + ko + 1];
    b.x = h[kk + 4 + ko];    b.y = h[kk + 4 + ko + 1];
    acc1 = __builtin_amdgcn_wmma_f32_16x16x4_f32(false, a, false, b, (short)0, acc1, false, false);
  }

  if ((lane & 15) == 0) {  // lane 0 -> rows row0..row0+7, lane 16 -> row0+8..row0+15
    const int base = row0 + (lane >> 4) * 8;
#pragma unroll
    for (int r = 0; r < 8; ++r)
      preact[base + r] = acc0[r] + acc1[r] + bi[base + r] + bh[base + r];
  }
}

// ---------------------------------------------------------------------------
// Streaming pass 1: cosine similarities + copy keys -> new_keys (NT hints).
// One wave per dict row (grid-stride); lane l owns 8 contiguous columns.
// sim[i] = dot(k_i, q) / ((||k_i||+eps) * (||q||+eps)),  q = h (pre-update).
// ---------------------------------------------------------------------------
__global__ __launch_bounds__(256)
void dnd_sim_copykeys(const float* __restrict__ keys, const float* __restrict__ h,
                      float* __restrict__ sim, float* __restrict__ newKeys) {
  const int lane = threadIdx.x & 31;
  const int gw   = (int)((blockIdx.x * blockDim.x + threadIdx.x) >> 5);
  const int nw   = (int)((gridDim.x * blockDim.x) >> 5);

  const v4f* h4 = (const v4f*)h;
  const v4f q0 = h4[lane * 2];
  const v4f q1 = h4[lane * 2 + 1];
  float qs = q0.x * q0.x + q0.y * q0.y + q0.z * q0.z + q0.w * q0.w +
             q1.x * q1.x + q1.y * q1.y + q1.z * q1.z + q1.w * q1.w;
  qs = waveReduceAdd(qs);
  const float qn = sqrtf(qs) + EPSF;

  for (int row = gw; row < DICT; row += nw) {
    const v4f* k4 = (const v4f*)(keys + (size_t)row * Hn);
    const v4f k0 = __builtin_nontemporal_load(&k4[lane * 2]);
    const v4f k1 = __builtin_nontemporal_load(&k4[lane * 2 + 1]);
    float d = k0.x * q0.x + k0.y * q0.y + k0.z * q0.z + k0.w * q0.w +
              k1.x * q1.x + k1.y * q1.y + k1.z * q1.z + k1.w * q1.w;
    float s = k0.x * k0.x + k0.y * k0.y + k0.z * k0.z + k0.w * k0.w +
              k1.x * k1.x + k1.y * k1.y + k1.z * k1.z + k1.w * k1.w;
    d = waveReduceAdd(d);
    s = waveReduceAdd(s);
    if (lane == 0) sim[row] = d / ((sqrtf(s) + EPSF) * qn);
    // copy-out (dest base misaligned by 1 element -> b32 stores)
    float* dst = newKeys + (size_t)row * Hn + lane * 8;
    __builtin_nontemporal_store(k0.x, &dst[0]);
    __builtin_nontemporal_store(k0.y, &dst[1]);
    __builtin_nontemporal_store(k0.z, &dst[2]);
    __builtin_nontemporal_store(k0.w, &dst[3]);
    __builtin_nontemporal_store(k1.x, &dst[4]);
    __builtin_nontemporal_store(k1.y, &dst[5]);
    __builtin_nontemporal_store(k1.z, &dst[6]);
    __builtin_nontemporal_store(k1.w, &dst[7]);
  }
}

// ---------------------------------------------------------------------------
// Streaming pass 2: w_i = exp(sim_i - 1) (cosine sim <= 1, so constant shift
// replaces the global max -> softmax unchanged). Accumulate sum(w) and
// sum(w*vals) via atomics; copy vals -> new_vals. Thread = one column.
// ---------------------------------------------------------------------------
__global__ __launch_bounds__(256)
void dnd_vals_weighted(const float* __restrict__ vals, const float* __restrict__ sim,
                       float* __restrict__ Vacc, float* __restrict__ Sacc,
                       float* __restrict__ newVals) {
  __shared__ float e[256];
  const int tid = threadIdx.x;
  const int r0  = blockIdx.x * 256;
  const int ch  = (DICT - r0 < 256) ? (DICT - r0) : 256;

  e[tid] = (tid < ch) ? expf(sim[r0 + tid] - 1.0f) : 0.0f;
  __syncthreads();

  float acc = 0.0f;
  for (int r = 0; r < ch; ++r) {
    const size_t idx = (size_t)(r0 + r) * Hn + tid;
    const float v = __builtin_nontemporal_load(&vals[idx]);
    acc = fmaf(e[r], v, acc);
    __builtin_nontemporal_store(v, &newVals[idx]);
  }
  atomicAdd(&Vacc[tid], acc);

  __syncthreads();
  for (int s = 128; s > 0; s >>= 1) {
    if (tid < s) e[tid] += e[tid + s];
    __syncthreads();
  }
  if (tid == 0) atomicAdd(Sacc, e[0]);
}

// ---------------------------------------------------------------------------
// Finalize: gates, c_t, h_t, dictionary row patch, policy softmax, value head.
// ---------------------------------------------------------------------------
__global__ __launch_bounds__(256)
void dnd_final(const float* __restrict__ preact, const float* __restrict__ cin,
               const float* __restrict__ Vacc, const float* __restrict__ Sacc,
               const float* __restrict__ Wa, const float* __restrict__ ba,
               const float* __restrict__ Wc, const float* __restrict__ bc,
               const int* __restrict__ widx,
               float* __restrict__ out, float* __restrict__ newKeys,
               float* __restrict__ newVals) {
  __shared__ float sh[Hn];
  __shared__ float sl[OUTS + 1];
  const int tid = threadIdx.x;

  const float f  = sigmoidf_(preact[tid]);
  const float i  = sigmoidf_(preact[Hn + tid]);
  const float o  = sigmoidf_(preact[2 * Hn + tid]);
  const float r  = sigmoidf_(preact[3 * Hn + tid]);
  const float cn = tanhf(preact[4 * Hn + tid]);
  const float m  = tanhf(Vacc[tid] / Sacc[0]);

  const float ct = f * cin[tid] + i * cn + r * m;
  const float ht = o * tanhf(ct);

  out[tid]      = ht;
  out[Hn + tid] = ct;

  const int wi = widx[0];
  newKeys[(size_t)wi * Hn + tid] = ht;
  newVals[(size_t)wi * Hn + tid] = ct;

  sh[tid] = ht;
  __syncthreads();

  const int wave = tid >> 5, lane = tid & 31;
  {  // wave w computes policy logit w
    const float* wr = Wa + wave * Hn;
    float p = 0.0f;
#pragma unroll
    for (int j = 0; j < 8; ++j) p = fmaf(wr[lane * 8 + j], sh[lane * 8 + j], p);
    p = waveReduceAdd(p);
    if (lane == 0) sl[wave] = p + ba[wave];
  }
  if (wave == 0) {  // value head
    float p = 0.0f;
#pragma unroll
    for (int j = 0; j < 8; ++j) p = fmaf(Wc[lane * 8 + j], sh[lane * 8 + j], p);
    p = waveReduceAdd(p);
    if (lane == 0) sl[OUTS] = p + bc[0];
  }
  __syncthreads();
  if (tid == 0) {
    float mx = sl[0];
    for (int q = 1; q < OUTS; ++q) mx = fmaxf(mx, sl[q]);
    float ee[OUTS], ssum = 0.0f;
    for (int q = 0; q < OUTS; ++q) { ee[q] = expf(sl[q] - mx); ssum += ee[q]; }
    for (int q = 0; q < OUTS; ++q) out[2 * Hn + q] = ee[q] / ssum;
    out[2 * Hn + OUTS] = sl[OUTS];
  }
}

extern "C" void kernel_launch(void* const* d_in, const int* in_sizes, int n_in,
                              void* d_out, int out_size, void* d_ws, size_t ws_size,
                              hipStream_t stream) {
  const float* x    = (const float*)d_in[0];
  const float* h    = (const float*)d_in[1];
  const float* c    = (const float*)d_in[2];
  const float* keys = (const float*)d_in[3];
  const float* vals = (const float*)d_in[4];
  const float* Wi   = (const float*)d_in[5];
  const float* bi   = (const float*)d_in[6];
  const float* Wh   = (const float*)d_in[7];
  const float* bh   = (const float*)d_in[8];
  const float* Wa   = (const float*)d_in[9];
  const float* ba   = (const float*)d_in[10];
  const float* Wc   = (const float*)d_in[11];
  const float* bc   = (const float*)d_in[12];
  const int*   widx = (const int*)d_in[13];

  float* out      = (float*)d_out;
  float* newKeys  = out + (2 * Hn + OUTS + 1);            // 521
  float* newVals  = newKeys + (size_t)DICT * Hn;

  float* ws     = (float*)d_ws;
  float* sim    = ws;                 // 100000 floats
  float* preact = ws + DICT;          // 1280 floats
  float* Vacc   = preact + ROWS;      // 256 floats
  float* Sacc   = Vacc + Hn;          // 1 float

  dnd_init<<<1, 256, 0, stream>>>(Vacc, Sacc);
  dnd_gates_wmma<<<ROWS / 16, 32, 0, stream>>>(x, h, Wi, bi, Wh, bh, preact);
  dnd_sim_copykeys<<<1024, 256, 0, stream>>>(keys, h, sim, newKeys);
  dnd_vals_weighted<<<(DICT + 255) / 256, 256, 0, stream>>>(vals, sim, Vacc, Sacc, newVals);
  dnd_final<<<1, 256, 0, stream>>>(preact, c, Vacc, Sacc, Wa, ba, Wc, bc, widx,
                                   out, newKeys, newVals);
}